// RATLayer_50010599194892
// MI455X (gfx1250) — compile-verified
//
#include <hip/hip_runtime.h>
#include <math.h>

#define D_MODEL 1024
#define N_HEADS 16
#define HEAD_DIM 64
#define K_RET 32
#define BANK 131072
#define BATCH 8
#define SEQ 2048
#define EPS_N 1e-12f

typedef __attribute__((ext_vector_type(2))) float v2f;
typedef __attribute__((ext_vector_type(8))) float v8f;

// ---------------------------------------------------------------------------
// 1) q[b][d] = mean_t x[b][t][d]
// ---------------------------------------------------------------------------
__global__ void rat_mean_kernel(const float* __restrict__ x, float* __restrict__ q) {
  int i = blockIdx.x * blockDim.x + threadIdx.x;   // 0 .. 8191
  int b = i >> 10, d = i & 1023;
  const float* xp = x + (size_t)b * SEQ * D_MODEL + d;
  float s = 0.f;
  for (int t = 0; t < SEQ; ++t) s += xp[(size_t)t * D_MODEL];
  q[i] = s * (1.0f / SEQ);
}

// ---------------------------------------------------------------------------
// 2) normalize q rows in place
// ---------------------------------------------------------------------------
__global__ void rat_qnorm_kernel(float* __restrict__ q) {
  int b = blockIdx.x;
  __shared__ float red[256];
  float s = 0.f;
  for (int d = threadIdx.x; d < D_MODEL; d += 256) {
    float v = q[b * D_MODEL + d];
    s += v * v;
  }
  red[threadIdx.x] = s;
  __syncthreads();
  for (int o = 128; o > 0; o >>= 1) {
    if (threadIdx.x < o) red[threadIdx.x] += red[threadIdx.x + o];
    __syncthreads();
  }
  float inv = 1.0f / fmaxf(sqrtf(red[0]), EPS_N);
  for (int d = threadIdx.x; d < D_MODEL; d += 256) q[b * D_MODEL + d] *= inv;
}

// ---------------------------------------------------------------------------
// 3) sims[b][n] = dot(qn[b], mem_keys[n]) / max(||mem_keys[n]||, eps)
//    one wave per key row; 8 batched dots + row norm; shuffle reduce.
//    Streams the full 536 MB key bank once — bandwidth-bound by design.
// ---------------------------------------------------------------------------
__global__ __launch_bounds__(256)
void rat_sims_kernel(const float* __restrict__ mem_keys,
                     const float* __restrict__ qn,
                     float* __restrict__ sims) {
  __shared__ float4 qlds[BATCH * 256];             // 8 x 1024 floats = 32 KB
  const float4* q4 = (const float4*)qn;
  for (int i = threadIdx.x; i < BATCH * 256; i += 256) qlds[i] = q4[i];
  __syncthreads();

  int wave = threadIdx.x >> 5, lane = threadIdx.x & 31;
  int n = blockIdx.x * 8 + wave;
  const float4* k4 = (const float4*)mem_keys + (size_t)n * 256;

  float nrm = 0.f;
  float dotb[BATCH];
#pragma unroll
  for (int b = 0; b < BATCH; ++b) dotb[b] = 0.f;

#pragma unroll
  for (int i = 0; i < 8; ++i) {
    float4 kv = k4[i * 32 + lane];
    nrm += kv.x * kv.x + kv.y * kv.y + kv.z * kv.z + kv.w * kv.w;
#pragma unroll
    for (int b = 0; b < BATCH; ++b) {
      float4 qv = qlds[b * 256 + i * 32 + lane];
      dotb[b] += kv.x * qv.x + kv.y * qv.y + kv.z * qv.z + kv.w * qv.w;
    }
  }
#pragma unroll
  for (int o = 16; o > 0; o >>= 1) {
    nrm += __shfl_xor(nrm, o, 32);
#pragma unroll
    for (int b = 0; b < BATCH; ++b) dotb[b] += __shfl_xor(dotb[b], o, 32);
  }
  if (lane == 0) {
    float inv = 1.0f / fmaxf(sqrtf(nrm), EPS_N);
#pragma unroll
    for (int b = 0; b < BATCH; ++b) sims[(size_t)b * BANK + n] = dotb[b] * inv;
  }
}

// ---------------------------------------------------------------------------
// 4) top-32 per batch: 32 argmax passes, LDS bitmask excludes winners.
//    Ties break to lower index (matches jax.lax.top_k ordering).
// ---------------------------------------------------------------------------
__global__ __launch_bounds__(1024)
void rat_topk_kernel(const float* __restrict__ sims, int* __restrict__ idx_out) {
  int b = blockIdx.x;
  __shared__ unsigned int mask[BANK / 32];         // 16 KB
  __shared__ float rv[1024];
  __shared__ int   ri[1024];
  for (int i = threadIdx.x; i < BANK / 32; i += 1024) mask[i] = 0u;
  __syncthreads();
  const float* s = sims + (size_t)b * BANK;

  for (int pass = 0; pass < K_RET; ++pass) {
    float bv = -INFINITY;
    int bi = BANK;
    for (int i = threadIdx.x; i < BANK; i += 1024) {
      if (mask[i >> 5] & (1u << (i & 31))) continue;
      float v = s[i];
      if (v > bv || (v == bv && i < bi)) { bv = v; bi = i; }
    }
    rv[threadIdx.x] = bv;
    ri[threadIdx.x] = bi;
    __syncthreads();
    for (int o = 512; o > 0; o >>= 1) {
      if (threadIdx.x < o) {
        float v2 = rv[threadIdx.x + o];
        int   i2 = ri[threadIdx.x + o];
        if (v2 > rv[threadIdx.x] || (v2 == rv[threadIdx.x] && i2 < ri[threadIdx.x])) {
          rv[threadIdx.x] = v2;
          ri[threadIdx.x] = i2;
        }
      }
      __syncthreads();
    }
    if (threadIdx.x == 0) {
      int w = ri[0];
      idx_out[b * K_RET + pass] = w;
      mask[w >> 5] |= (1u << (w & 31));
    }
    __syncthreads();
  }
}

// ---------------------------------------------------------------------------
// 5) retrieved[r][:] = mem_values[idx[r]][:]   (r = b*32 + j)
// ---------------------------------------------------------------------------
__global__ void rat_gather_kernel(const float* __restrict__ mem_values,
                                  const int* __restrict__ idx,
                                  float* __restrict__ retrieved) {
  int r = blockIdx.x;
  int src = idx[r];
  const float4* in = (const float4*)mem_values + (size_t)src * 256;
  float4* outp = (float4*)retrieved + (size_t)r * 256;
  outp[threadIdx.x] = in[threadIdx.x];
}

// ---------------------------------------------------------------------------
// 6) fp32 WMMA GEMM: C(MxN) = A(MxK) * B(KxN), all row-major.
//    64x64 tile per block, 4 waves, each wave owns a 16-row strip (4 WMMA accs).
//    Double-buffered LDS staging via CDNA5 async global->LDS loads
//    (GLOBAL_LOAD_ASYNC_TO_LDS_B128, ASYNCcnt): next K-tile's HBM traffic
//    overlaps the current tile's WMMA burst; no VGPR round-trip for staging.
//    V_WMMA_F32_16X16X4_F32 fragment layouts per ISA 7.12.2.
// ---------------------------------------------------------------------------
#define TM 64
#define TN 64
#define TK 32
#define APITCH 36
#define BPITCH 68

__global__ __launch_bounds__(128)
void rat_wmma_gemm_kernel(const float* __restrict__ A, const float* __restrict__ B,
                          float* __restrict__ C, int M, int N, int K) {
  __shared__ float As[2][TM * APITCH];   // 2 x 9216 B
  __shared__ float Bs[2][TK * BPITCH];   // 2 x 8704 B
  int m0 = blockIdx.y * TM, n0 = blockIdx.x * TN;
  int tid = threadIdx.x, wave = tid >> 5, lane = tid & 31;
  int lhalf = lane >> 4, l16 = lane & 15;

  // Per-thread staging coordinates (4 x b128 for A, 4 x b128 for B per tile).
  int aRow[4], aCol[4], bRow[4], bCol[4];
  unsigned aLds[2][4], bLds[2][4];
#pragma unroll
  for (int i = 0; i < 4; ++i) {
    int lin = i * 128 + tid;                 // 0..511
    aRow[i] = lin >> 3; aCol[i] = (lin & 7) << 2;    // 64 rows x 8 float4
    bRow[i] = lin >> 4; bCol[i] = (lin & 15) << 2;   // 32 rows x 16 float4
#pragma unroll
    for (int buf = 0; buf < 2; ++buf) {
      aLds[buf][i] = (unsigned)(uintptr_t)&As[buf][aRow[i] * APITCH + aCol[i]];
      bLds[buf][i] = (unsigned)(uintptr_t)&Bs[buf][bRow[i] * BPITCH + bCol[i]];
    }
  }

  // Issue one K-tile's async global->LDS loads (8 x b128 per thread).
  auto stage = [&](int buf, int k0) {
#pragma unroll
    for (int i = 0; i < 4; ++i) {
      unsigned ga = (unsigned)(((size_t)(m0 + aRow[i]) * K + (k0 + aCol[i])) * sizeof(float));
      asm volatile("global_load_async_to_lds_b128 %0, %1, %2"
                   :: "v"(aLds[buf][i]), "v"(ga), "s"(A) : "memory");
      unsigned gb = (unsigned)(((size_t)(k0 + bRow[i]) * N + (n0 + bCol[i])) * sizeof(float));
      asm volatile("global_load_async_to_lds_b128 %0, %1, %2"
                   :: "v"(bLds[buf][i]), "v"(gb), "s"(B) : "memory");
    }
  };

  v8f acc[4];
#pragma unroll
  for (int t = 0; t < 4; ++t)
#pragma unroll
    for (int j = 0; j < 8; ++j) acc[t][j] = 0.f;

  stage(0, 0);
  int nstages = K / TK;
  for (int st = 0; st < nstages; ++st) {
    int cur = st & 1;
    if (st + 1 < nstages) {
      stage(cur ^ 1, (st + 1) * TK);
      // 8 next-tile loads may remain outstanding; current tile's 8 are done.
      asm volatile("s_wait_asynccnt 0x8" ::: "memory");
    } else {
      asm volatile("s_wait_asynccnt 0x0" ::: "memory");
    }
    __syncthreads();

    const float* Asb = &As[cur][0];
    const float* Bsb = &Bs[cur][0];
#pragma unroll
    for (int kk = 0; kk < TK; kk += 4) {
      int am = wave * 16 + l16;
      int ak = kk + lhalf * 2;
      v2f a;
      a.x = Asb[am * APITCH + ak];
      a.y = Asb[am * APITCH + ak + 1];
#pragma unroll
      for (int t = 0; t < 4; ++t) {
        int bn = t * 16 + l16;
        int bk = kk + lhalf * 2;
        v2f bb;
        bb.x = Bsb[bk * BPITCH + bn];
        bb.y = Bsb[(bk + 1) * BPITCH + bn];
        acc[t] = __builtin_amdgcn_wmma_f32_16x16x4_f32(
            /*neg_a=*/false, a, /*neg_b=*/false, bb,
            /*c_mod=*/(short)0, acc[t], /*reuse_a=*/false, /*reuse_b=*/false);
      }
    }
    __syncthreads();
  }

  // store C per ISA 16x16 f32 C/D layout
#pragma unroll
  for (int t = 0; t < 4; ++t) {
    int col = n0 + t * 16 + l16;
#pragma unroll
    for (int j = 0; j < 8; ++j) {
      int row = m0 + wave * 16 + j + lhalf * 8;
      C[(size_t)row * N + col] = acc[t][j];
    }
  }
}

// ---------------------------------------------------------------------------
// 7) cross-attention over the 32 retrieved rows, per (b, h).
//    K/V head slices cached in LDS (pitch 65 => conflict-free column reads).
//    One wave per query row t: lane j = key j; shfl softmax; shfl-broadcast AV.
// ---------------------------------------------------------------------------
#define KPITCH 65
__global__ __launch_bounds__(256)
void rat_attn_kernel(const float* __restrict__ Q, const float* __restrict__ Kp,
                     const float* __restrict__ Vp, float* __restrict__ ctx) {
  int tchunk = blockIdx.x;   // 0..7 -> 256 rows each
  int h = blockIdx.y;        // 0..15
  int b = blockIdx.z;        // 0..7
  __shared__ float Kl[K_RET * KPITCH];
  __shared__ float Vl[K_RET * KPITCH];
  for (int i = threadIdx.x; i < K_RET * HEAD_DIM; i += 256) {
    int r = i >> 6, d = i & 63;
    size_t g = (size_t)(b * K_RET + r) * D_MODEL + h * HEAD_DIM + d;
    Kl[r * KPITCH + d] = Kp[g];
    Vl[r * KPITCH + d] = Vp[g];
  }
  __syncthreads();

  int wave = threadIdx.x >> 5, lane = threadIdx.x & 31;
  const float scale = 0.125f;  // 1/sqrt(64)

  for (int it = 0; it < 32; ++it) {
    int t = tchunk * 256 + it * 8 + wave;
    size_t qoff = (size_t)(b * SEQ + t) * D_MODEL + h * HEAD_DIM;

    float s = 0.f;
#pragma unroll
    for (int d = 0; d < HEAD_DIM; ++d)
      s += Q[qoff + d] * Kl[lane * KPITCH + d];   // Q read is wave-uniform
    s *= scale;

    float m = s;
#pragma unroll
    for (int o = 16; o > 0; o >>= 1) m = fmaxf(m, __shfl_xor(m, o, 32));
    float p = __expf(s - m);
    float sum = p;
#pragma unroll
    for (int o = 16; o > 0; o >>= 1) sum += __shfl_xor(sum, o, 32);
    p /= sum;

    float a0 = 0.f, a1 = 0.f;
#pragma unroll
    for (int j = 0; j < K_RET; ++j) {
      float pj = __shfl(p, j, 32);
      a0 += pj * Vl[j * KPITCH + lane];
      a1 += pj * Vl[j * KPITCH + lane + 32];
    }
    ctx[qoff + lane] = a0;
    ctx[qoff + lane + 32] = a1;
  }
}

// ---------------------------------------------------------------------------
// 8) out = x + sigmoid(gate_logit) * ctxWo
// ---------------------------------------------------------------------------
__global__ void rat_residual_kernel(const float* __restrict__ x,
                                    const float* __restrict__ c,
                                    const float* __restrict__ gate_logit,
                                    float* __restrict__ out, int n) {
  int i = blockIdx.x * blockDim.x + threadIdx.x;
  if (i < n) {
    float g = 1.0f / (1.0f + __expf(-gate_logit[0]));
    out[i] = x[i] + g * c[i];
  }
}

// ---------------------------------------------------------------------------
extern "C" void kernel_launch(void* const* d_in, const int* in_sizes, int n_in,
                              void* d_out, int out_size, void* d_ws, size_t ws_size,
                              hipStream_t stream) {
  const float* x          = (const float*)d_in[0];
  const float* mem_keys   = (const float*)d_in[1];
  const float* mem_values = (const float*)d_in[2];
  const float* Wq         = (const float*)d_in[3];
  const float* Wk         = (const float*)d_in[4];
  const float* Wv         = (const float*)d_in[5];
  const float* Wo         = (const float*)d_in[6];
  const float* gate       = (const float*)d_in[7];
  float* out = (float*)d_out;

  float* ws = (float*)d_ws;
  float* qn        = ws; ws += (size_t)BATCH * D_MODEL;          // 8K
  float* sims      = ws; ws += (size_t)BATCH * BANK;             // 1M
  float* retrieved = ws; ws += (size_t)BATCH * K_RET * D_MODEL;  // 256K
  float* Qb        = ws; ws += (size_t)BATCH * SEQ * D_MODEL;    // 16M
  float* Kb        = ws; ws += (size_t)BATCH * K_RET * D_MODEL;  // 256K
  float* Vb        = ws; ws += (size_t)BATCH * K_RET * D_MODEL;  // 256K
  float* ctx       = ws; ws += (size_t)BATCH * SEQ * D_MODEL;    // 16M
  float* ctx2      = ws; ws += (size_t)BATCH * SEQ * D_MODEL;    // 16M
  int* idx = (int*)ws;

  rat_mean_kernel<<<BATCH * D_MODEL / 256, 256, 0, stream>>>(x, qn);
  rat_qnorm_kernel<<<BATCH, 256, 0, stream>>>(qn);
  rat_sims_kernel<<<BANK / 8, 256, 0, stream>>>(mem_keys, qn, sims);
  rat_topk_kernel<<<BATCH, 1024, 0, stream>>>(sims, idx);
  rat_gather_kernel<<<BATCH * K_RET, 256, 0, stream>>>(mem_values, idx, retrieved);

  dim3 gBig(D_MODEL / TN, (BATCH * SEQ) / TM);     // (16, 256)
  dim3 gSmall(D_MODEL / TN, (BATCH * K_RET) / TM); // (16, 4)
  rat_wmma_gemm_kernel<<<gBig,   128, 0, stream>>>(x, Wq, Qb, BATCH * SEQ, D_MODEL, D_MODEL);
  rat_wmma_gemm_kernel<<<gSmall, 128, 0, stream>>>(retrieved, Wk, Kb, BATCH * K_RET, D_MODEL, D_MODEL);
  rat_wmma_gemm_kernel<<<gSmall, 128, 0, stream>>>(retrieved, Wv, Vb, BATCH * K_RET, D_MODEL, D_MODEL);

  dim3 gAttn(SEQ / 256, N_HEADS, BATCH);           // (8, 16, 8)
  rat_attn_kernel<<<gAttn, 256, 0, stream>>>(Qb, Kb, Vb, ctx);

  rat_wmma_gemm_kernel<<<gBig, 128, 0, stream>>>(ctx, Wo, ctx2, BATCH * SEQ, D_MODEL, D_MODEL);

  int n = BATCH * SEQ * D_MODEL;
  rat_residual_kernel<<<n / 256, 256, 0, stream>>>(x, ctx2, gate, out, n);
}